// Reservoir_80788334838317
// MI455X (gfx1250) — compile-verified
//
#include <hip/hip_runtime.h>
#include <cmath>

typedef __attribute__((ext_vector_type(2))) float v2f;
typedef __attribute__((ext_vector_type(8))) float v8f;

#define L_STEPS 2048
#define NBATCH  32
#define HID     1024
#define INS     128
#define NWG     16
#define COLS_PER_WG (HID / NWG)   // 64 output columns per workgroup
#define THREADS 256               // 8 waves of 32

// ---------------------------------------------------------------------------
// Grid barrier: generation counter + arrive count. count returns to 0 after
// every barrier; gen grows monotonically (value irrelevant, only change
// detection), so repeated graph replays are deterministic.
// ---------------------------------------------------------------------------
__global__ void init_bar_kernel(unsigned* bar) { bar[0] = 0u; }

__device__ inline void grid_barrier(unsigned* count, unsigned* gen) {
  __syncthreads();
  __threadfence();  // release: make this WG's stores device-visible
  if (threadIdx.x == 0) {
    unsigned g = __atomic_load_n(gen, __ATOMIC_RELAXED);
    unsigned prev = __atomic_fetch_add(count, 1u, __ATOMIC_ACQ_REL);
    if (prev == NWG - 1) {
      __atomic_store_n(count, 0u, __ATOMIC_RELAXED);
      __atomic_fetch_add(gen, 1u, __ATOMIC_RELEASE);
    } else {
      while (__atomic_load_n(gen, __ATOMIC_ACQUIRE) == g) {
        __builtin_amdgcn_s_sleep(2);
      }
    }
  }
  __syncthreads();
  __threadfence();  // acquire: invalidate near caches before reading peers' h
}

// ---------------------------------------------------------------------------
// Persistent reservoir kernel. Each WG owns 64 hidden columns; W_h / W_in
// slices live in LDS pre-transposed as [k/4][n][4] so a lane's B fragment
// (2 consecutive K) is one 64-bit LDS load with no bank conflicts.
// ---------------------------------------------------------------------------
__global__ __launch_bounds__(THREADS, 1)
void reservoir_step_kernel(const float* __restrict__ x,       // (L,N,I)
                           const float* __restrict__ h_init,  // (N,H)
                           const float* __restrict__ W_in,    // (H,I)
                           const float* __restrict__ bias,    // (H)
                           const float* __restrict__ W_h,     // (H,H)
                           float* __restrict__ out,           // (L,N,H)
                           unsigned* __restrict__ bar) {
  __shared__ float lds_wh [HID * COLS_PER_WG];   // 256 KB  [kb][n][4]
  __shared__ float lds_win[INS * COLS_PER_WG];   //  32 KB  [kb][n][4]
  __shared__ float lds_bias[COLS_PER_WG];

  const int tid      = threadIdx.x;
  const int col_base = blockIdx.x * COLS_PER_WG;

  // ---- one-time LDS fill (coalesced global reads, transposed LDS writes) ----
  for (int idx = tid; idx < COLS_PER_WG * HID; idx += THREADS) {
    int n = idx >> 10;            // / HID
    int k = idx & (HID - 1);
    float w = W_h[(size_t)(col_base + n) * HID + k];
    lds_wh[(k >> 2) * (COLS_PER_WG * 4) + n * 4 + (k & 3)] = w;
  }
  for (int idx = tid; idx < COLS_PER_WG * INS; idx += THREADS) {
    int n = idx >> 7;             // / INS
    int k = idx & (INS - 1);
    float w = W_in[(size_t)(col_base + n) * INS + k];
    lds_win[(k >> 2) * (COLS_PER_WG * 4) + n * 4 + (k & 3)] = w;
  }
  if (tid < COLS_PER_WG) lds_bias[tid] = bias[col_base + tid];
  __syncthreads();

  // ---- per-lane fragment geometry (16x16x4 f32 WMMA layouts) ----
  const int lane     = tid & 31;
  const int wave     = tid >> 5;
  const int row_base = (wave >> 2) * 16;   // M tile: 0 or 16 (batch rows)
  const int n_sub    = (wave & 3) * 16;    // N tile within the 64-col slice
  const int nl       = lane & 15;
  const int hi       = lane >> 4;          // 0: K+{0,1}   1: K+{2,3}
  const int kdelta   = hi * 2;

  const int   m      = row_base + nl;                          // A row
  const float* wh_b  = &lds_wh [n_sub * 4 + nl * 4 + kdelta];  // B base (W_h)
  const float* win_b = &lds_win[n_sub * 4 + nl * 4 + kdelta];  // B base (W_in)
  const float  bval  = lds_bias[n_sub + nl];

  for (int t = 0; t < L_STEPS; ++t) {
    // previous hidden state row lives in d_out (states[t-1]); h_init at t==0
    const float* hrow = (t == 0)
        ? (h_init + (size_t)m * HID)
        : (out + ((size_t)(t - 1) * NBATCH + m) * HID);
    const float* xrow = x + ((size_t)t * NBATCH + m) * INS;

    v8f c0 = {}, c1 = {};

    // h @ W_h^T : K = 1024, dual accumulator chains for pipelining
    #pragma unroll 4
    for (int kb = 0; kb < HID / 4; kb += 2) {
      v2f a0 = *(const v2f*)(hrow + kb * 4 + kdelta);
      v2f b0 = *(const v2f*)(wh_b + kb * (COLS_PER_WG * 4));
      c0 = __builtin_amdgcn_wmma_f32_16x16x4_f32(false, a0, false, b0,
                                                 (short)0, c0, false, false);
      v2f a1 = *(const v2f*)(hrow + (kb + 1) * 4 + kdelta);
      v2f b1 = *(const v2f*)(wh_b + (kb + 1) * (COLS_PER_WG * 4));
      c1 = __builtin_amdgcn_wmma_f32_16x16x4_f32(false, a1, false, b1,
                                                 (short)0, c1, false, false);
    }
    // x[t] @ W_in^T : K = 128 (fused input projection)
    #pragma unroll 4
    for (int kb = 0; kb < INS / 4; kb += 2) {
      v2f a0 = *(const v2f*)(xrow + kb * 4 + kdelta);
      v2f b0 = *(const v2f*)(win_b + kb * (COLS_PER_WG * 4));
      c0 = __builtin_amdgcn_wmma_f32_16x16x4_f32(false, a0, false, b0,
                                                 (short)0, c0, false, false);
      v2f a1 = *(const v2f*)(xrow + (kb + 1) * 4 + kdelta);
      v2f b1 = *(const v2f*)(win_b + (kb + 1) * (COLS_PER_WG * 4));
      c1 = __builtin_amdgcn_wmma_f32_16x16x4_f32(false, a1, false, b1,
                                                 (short)0, c1, false, false);
    }

    // bias + tanh + store: C elem j -> row = row_base + hi*8 + j, col fixed
    float* ocol = out + (size_t)t * NBATCH * HID + (size_t)(col_base + n_sub + nl);
    #pragma unroll
    for (int j = 0; j < 8; ++j) {
      float v = tanhf(c0[j] + c1[j] + bval);
      int row = row_base + hi * 8 + j;
      ocol[(size_t)row * HID] = v;
    }

    grid_barrier(bar, bar + 1);
  }
}

// ---------------------------------------------------------------------------
extern "C" void kernel_launch(void* const* d_in, const int* in_sizes, int n_in,
                              void* d_out, int out_size, void* d_ws, size_t ws_size,
                              hipStream_t stream) {
  const float* x      = (const float*)d_in[0];
  const float* h_init = (const float*)d_in[1];
  const float* W_in   = (const float*)d_in[2];
  const float* bias   = (const float*)d_in[3];
  const float* W_h    = (const float*)d_in[4];
  float* out          = (float*)d_out;
  unsigned* bar       = (unsigned*)d_ws;   // bar[0]=count, bar[1]=gen

  init_bar_kernel<<<1, 1, 0, stream>>>(bar);
  reservoir_step_kernel<<<NWG, THREADS, 0, stream>>>(x, h_init, W_in, bias,
                                                     W_h, out, bar);
}